// RGBChannelAttention_7387343749841
// MI455X (gfx1250) — compile-verified
//
#include <hip/hip_runtime.h>

// MI455X / gfx1250, wave32. Phase 1: GAP over H*W=196 per (n,c,t) done as
// X(rows x 196) * ones(196)/196 via V_WMMA_F32_16X16X4_F32 (f32 end-to-end).
// Phase 2: 5-tap conv over channel axis + sigmoid (tiny, ~1MB traffic).

typedef __attribute__((ext_vector_type(2))) float v2f;
typedef __attribute__((ext_vector_type(8))) float v8f;

#define GAP_HW     196   // 14*14
#define GAP_KSTEPS 49    // 196 / 4

// One wave reduces 16 rows (rows = flattened (n,c,t), contiguous 196 floats each).
// B-tile layout for 16x16x4 f32 WMMA (wave32):
//   VGPR0: lanes 0-15 -> K=0, lanes 16-31 -> K=2
//   VGPR1: lanes 0-15 -> K=1, lanes 16-31 -> K=3
// => each lane loads 2 consecutive floats at hw = 4*kk + 2*(lane>>4): one b64 load.
__global__ void __launch_bounds__(256)
gap_wmma_kernel(const float* __restrict__ x, float* __restrict__ gap) {
    const int lane = threadIdx.x & 31;
    const int wave = (int)((blockIdx.x * blockDim.x + threadIdx.x) >> 5);
    const int col  = lane & 15;   // N index (which of the 16 rows this lane carries in B)
    const int half = lane >> 4;   // K offset selector: 0 -> K=0,1 ; 1 -> K=2,3
    const long row = (long)wave * 16 + col;

    const float* p = x + row * (long)GAP_HW + 2 * half;

    v2f a;                 // A matrix: every element = 1/196 (fold the mean into A)
    a[0] = 1.0f / 196.0f;
    a[1] = 1.0f / 196.0f;

    v8f acc = {};          // C/D accumulator, 16x16 f32
    #pragma unroll 7
    for (int kk = 0; kk < GAP_KSTEPS; ++kk) {
        v2f b = *reinterpret_cast<const v2f*>(p + kk * 4);  // global_load_b64, 8B aligned
        // D = A(16x4, const 1/196) x B(4x16, data) + C  -> every D row = means of the 16 rows
        acc = __builtin_amdgcn_wmma_f32_16x16x4_f32(
            /*neg_a=*/false, a, /*neg_b=*/false, b,
            /*c_mod=*/(short)0, acc, /*reuse_a=*/false, /*reuse_b=*/false);
    }

    // D layout: VGPR0, lanes 0-15 = (M=0, N=lane) -> the 16 means.
    if (lane < 16) {
        gap[row] = acc[0];
    }
}

// gap is flat (N, C, T): index i = (n*C + c)*T + t. Conv over c => neighbors at i +- j*T.
// jax/torch conv = cross-correlation: y[c] = sum_j w[j] * gap[c + j - 2], zero-padded.
__global__ void __launch_bounds__(256)
conv_sigmoid_kernel(const float* __restrict__ gap, const float* __restrict__ w,
                    float* __restrict__ out, int C, int T, int total) {
    const int i = blockIdx.x * blockDim.x + threadIdx.x;
    if (i >= total) return;

    const int c = (i / T) % C;

    float s = 0.0f;
    #pragma unroll
    for (int j = 0; j < 5; ++j) {
        const int cc = c + j - 2;
        if (cc >= 0 && cc < C) {
            s += w[j] * gap[i + (j - 2) * T];
        }
    }
    out[i] = 1.0f / (1.0f + __expf(-s));
}

extern "C" void kernel_launch(void* const* d_in, const int* in_sizes, int n_in,
                              void* d_out, int out_size, void* d_ws, size_t ws_size,
                              hipStream_t stream) {
    const float* x = (const float*)d_in[0];   // (8, 512, 32, 14, 14) f32
    const float* w = (const float*)d_in[1];   // (1, 1, 5) f32
    float* out = (float*)d_out;               // (8, 512, 32, 1, 1) f32
    float* gap = (float*)d_ws;                // 131072 f32 scratch (512 KB)

    const int N = 8, C = 512, T = 32;
    const int rows  = N * C * T;              // 131072 (multiple of 16)
    const int block = 256;                    // 8 wave32 waves per block

    // Phase 1: 16 rows per wave -> rows/16 waves -> *32 threads.
    const int waves = rows / 16;              // 8192
    const int grid1 = (waves * 32) / block;   // 1024 blocks
    gap_wmma_kernel<<<grid1, block, 0, stream>>>(x, gap);

    // Phase 2: one thread per output element.
    const int grid2 = (rows + block - 1) / block;
    conv_sigmoid_kernel<<<grid2, block, 0, stream>>>(gap, w, out, C, T, rows);
}